// SpatialTransformer_36524401885483
// MI455X (gfx1250) — compile-verified
//
#include <hip/hip_runtime.h>

// Problem constants from the reference: B, H, W = 32, 768, 768
constexpr int kB = 32;
constexpr int kH = 768;
constexpr int kW = 768;
constexpr int kHW = kH * kW;

// Native clang vector type (required by __builtin_nontemporal_load/store).
typedef float v4f __attribute__((ext_vector_type(4)));

// Bilinear backward-warp (spatial transformer sampler), channel count = 1.
// Grid shape carries the index decomposition (no integer division):
//   blockIdx.z = batch, blockIdx.y = row, threadIdx.x*4 = column.
// Each thread produces 4 consecutive output pixels along W:
//   - flow read as two b128 non-temporal loads (streamed, read-once)
//   - 16 scattered b32 gathers from src (src = 75.5 MB, fits in 192 MB L2)
//   - result written as one b128 non-temporal store
__global__ __launch_bounds__(192) void SpatialTransformer_warp_kernel(
    const float* __restrict__ src,   // [B,H,W,1]
    const float* __restrict__ flow,  // [B,H,W,2]
    float* __restrict__ out)         // [B,H,W,1]
{
    const int x = threadIdx.x << 2;          // 0..764, multiple of 4
    const int y = blockIdx.y;                // 0..767
    const int b = blockIdx.z;                // 0..31

    const int base = (b * kH + y) * kW + x;  // flat pixel index (< 2^25)

    const float* __restrict__ img = src + (long long)b * kHW;

    // Flow for 4 pixels = 8 floats, 32-byte aligned (base is a multiple of 4).
    const v4f* fptr = (const v4f*)(flow + 2ll * base);
    const v4f f01 = __builtin_nontemporal_load(fptr);      // (fx0,fy0,fx1,fy1)
    const v4f f23 = __builtin_nontemporal_load(fptr + 1);  // (fx2,fy2,fx3,fy3)

    const float fx[4] = {f01.x, f01.z, f23.x, f23.z};
    const float fy[4] = {f01.y, f01.w, f23.y, f23.w};

    float res[4];
#pragma unroll
    for (int j = 0; j < 4; ++j) {
        const float gx = (float)(x + j) + fx[j];
        const float gy = (float)y + fy[j];

        const float x0f = floorf(gx);
        const float y0f = floorf(gy);
        const float x1f = x0f + 1.0f;
        const float y1f = y0f + 1.0f;

        // Weights from UNCLIPPED corners (matches reference).
        const float wxh = x1f - gx;   // weight toward x0
        const float wxl = gx - x0f;   // weight toward x1
        const float wyh = y1f - gy;   // weight toward y0
        const float wyl = gy - y0f;   // weight toward y1
        const float wa = wxh * wyh;
        const float wb = wxl * wyh;
        const float wc = wxh * wyl;
        const float wd = wxl * wyl;

        // Clip as float, then truncate to int (exact: values are integral >= 0).
        const int x0 = (int)fminf(fmaxf(x0f, 0.0f), (float)(kW - 1));
        const int x1 = (int)fminf(fmaxf(x1f, 0.0f), (float)(kW - 1));
        const int y0 = (int)fminf(fmaxf(y0f, 0.0f), (float)(kH - 1));
        const int y1 = (int)fminf(fmaxf(y1f, 0.0f), (float)(kH - 1));

        const int r0 = y0 * kW;
        const int r1 = y1 * kW;
        const float va = img[r0 + x0];
        const float vb = img[r0 + x1];
        const float vc = img[r1 + x0];
        const float vd = img[r1 + x1];

        res[j] = wa * va + wb * vb + wc * vc + wd * vd;
    }

    v4f o;
    o.x = res[0]; o.y = res[1]; o.z = res[2]; o.w = res[3];
    __builtin_nontemporal_store(o, (v4f*)(out + base));
}

extern "C" void kernel_launch(void* const* d_in, const int* in_sizes, int n_in,
                              void* d_out, int out_size, void* d_ws, size_t ws_size,
                              hipStream_t stream) {
    (void)in_sizes; (void)n_in; (void)out_size; (void)d_ws; (void)ws_size;

    const float* src  = (const float*)d_in[0];  // [B,H,W,1] float32
    const float* flow = (const float*)d_in[1];  // [B,H,W,2] float32
    float*       out  = (float*)d_out;          // [B,H,W,1] float32

    // One block per image row: 192 threads * 4 px = 768 px = W.
    // 6 wave32 per block; 768 rows * 32 batches = 24576 blocks.
    const dim3 grid(1, kH, kB);
    const dim3 block(192, 1, 1);

    SpatialTransformer_warp_kernel<<<grid, block, 0, stream>>>(src, flow, out);
}